// RandomCutoff_54984171323497
// MI455X (gfx1250) — compile-verified
//
#include <hip/hip_runtime.h>
#include <stdint.h>

// ---------------- problem constants ----------------
#define TIME_MASK_MAX_LEN 10
#define TIME_MASK_STEP    40
#define FREQ_MASK_MAX_LEN 35

#define DIM_B 32
#define DIM_C 7
#define DIM_T 960
#define DIM_F 128
#define CM    4            // C - 3 masked channels
#define NB    24           // T / TIME_MASK_STEP (exact: 960/40)
#define NRECT (CM * NB)    // 96 rectangles

#define N_ELEM  27525120u          // 32*7*960*128
#define N_VEC4  (N_ELEM / 4u)      // 6,881,280 float4
#define LOG_EPS (-36.04365338911715f)  // log(np.finfo(float64).eps), rounded to f32

typedef float vf4 __attribute__((ext_vector_type(4)));

struct __align__(16) Rect { int ts, te, fs, fe; };

// ---------------- Threefry-2x32 (JAX-compatible) ----------------
__device__ __forceinline__ uint32_t rotl32(uint32_t v, int r) {
    return (v << r) | (v >> (32 - r));
}

__device__ __forceinline__ void tf4(uint32_t &x0, uint32_t &x1,
                                    int r0, int r1, int r2, int r3) {
    x0 += x1; x1 = rotl32(x1, r0); x1 ^= x0;
    x0 += x1; x1 = rotl32(x1, r1); x1 ^= x0;
    x0 += x1; x1 = rotl32(x1, r2); x1 ^= x0;
    x0 += x1; x1 = rotl32(x1, r3); x1 ^= x0;
}

__device__ void threefry2x32(uint32_t k0, uint32_t k1,
                             uint32_t x0, uint32_t x1,
                             uint32_t &y0, uint32_t &y1) {
    uint32_t k2 = k0 ^ k1 ^ 0x1BD11BDAu;
    x0 += k0; x1 += k1;
    tf4(x0, x1, 13, 15, 26, 6);   x0 += k1; x1 += k2 + 1u;
    tf4(x0, x1, 17, 29, 16, 24);  x0 += k2; x1 += k0 + 2u;
    tf4(x0, x1, 13, 15, 26, 6);   x0 += k0; x1 += k1 + 3u;
    tf4(x0, x1, 17, 29, 16, 24);  x0 += k1; x1 += k2 + 4u;
    tf4(x0, x1, 13, 15, 26, 6);   x0 += k2; x1 += k0 + 5u;
    y0 = x0; y1 = x1;
}

// uniform[0,1): (bits >> 9) | 0x3f800000, bitcast, minus 1  (JAX _uniform)
__device__ __forceinline__ float bits_to_unit(uint32_t bits) {
    return __uint_as_float((bits >> 9) | 0x3f800000u) - 1.0f;
}

// ---------------- setup kernel: derive the 96 rectangles + apply flag ------
// jax.random.key(42) -> (0,42); partitionable counter scheme:
//   split(key)[i]        = threefry(key, (0, i))         (both output words)
//   random_bits32(key)_i = threefry(key, (0, i)).word1
__global__ void __launch_bounds__(128)
rng_setup_kernel(int* __restrict__ ws) {
    const int j = threadIdx.x;

    uint32_t a0, a1;
    uint32_t ka0, ka1, km0, km1;
    threefry2x32(0u, 42u, 0u, 0u, ka0, ka1);   // k_apply = split(root)[0]
    threefry2x32(0u, 42u, 0u, 1u, km0, km1);   // k_mask  = split(root)[1]

    uint32_t ktl0, ktl1, kts0, kts1, kfl0, kfl1, kfs0, kfs1;
    threefry2x32(km0, km1, 0u, 0u, ktl0, ktl1);  // k_tl
    threefry2x32(km0, km1, 0u, 1u, kts0, kts1);  // k_ts
    threefry2x32(km0, km1, 0u, 2u, kfl0, kfl1);  // k_fl
    threefry2x32(km0, km1, 0u, 3u, kfs0, kfs1);  // k_fs

    if (j < NRECT) {
        uint32_t btl, bts, bfl, bfs;
        threefry2x32(ktl0, ktl1, 0u, (uint32_t)j, a0, btl);
        threefry2x32(kts0, kts1, 0u, (uint32_t)j, a0, bts);
        threefry2x32(kfl0, kfl1, 0u, (uint32_t)j, a0, bfl);
        threefry2x32(kfs0, kfs1, 0u, (uint32_t)j, a0, bfs);

        // randint(0, 10): bits % 10 ; randint(0, 35): bits % 35  (JAX _randint)
        int tlen = (int)(btl % (uint32_t)TIME_MASK_MAX_LEN);
        int flen = (int)(bfl % (uint32_t)FREQ_MASK_MAX_LEN);
        int tstart = (int)floorf(bits_to_unit(bts) * (float)(TIME_MASK_STEP - tlen));
        int fstart = (int)floorf(bits_to_unit(bfs) * (float)(DIM_F - flen));

        int4 r;
        r.x = tstart; r.y = tstart + tlen;
        r.z = fstart; r.w = fstart + flen;
        ((int4*)ws)[j] = r;
    }
    if (j == NRECT) {
        uint32_t b0, b1;
        threefry2x32(ka0, ka1, 0u, 0u, b0, b1);     // scalar uniform draw
        ws[4 * NRECT] = (bits_to_unit(b1) < 0.3f) ? 1 : 0;
    }
}

// ---------------- streaming mask-apply kernel ----------------
#define BLOCK 256
#define ITEMS 4

__global__ void __launch_bounds__(BLOCK)
cutoff_apply_kernel(const vf4* __restrict__ x, vf4* __restrict__ out,
                    const int* __restrict__ ws) {
    __shared__ Rect s_rect[NRECT];
    __shared__ int  s_apply;

    const int tid = threadIdx.x;

    // gfx1250 async DMA: stage the 96 rectangle descriptors into LDS without
    // touching VGPRs (one b128 per lane, waves 0-2 fully active).
    if (tid < NRECT) {
        uint32_t lds_off = (uint32_t)(uintptr_t)&s_rect[tid];   // low 32b of a
        uint64_t gaddr   = (uint64_t)(uintptr_t)(ws + 4 * tid); // shared ptr ==
        asm volatile("global_load_async_to_lds_b128 %0, %1, off" // LDS offset
                     :: "v"(lds_off), "v"(gaddr)
                     : "memory");
        asm volatile("s_wait_asynccnt 0x0" ::: "memory");
    }
    if (tid == NRECT) s_apply = ws[4 * NRECT];
    __syncthreads();

    const int apply = s_apply;
    uint32_t idx = (uint32_t)blockIdx.x * (BLOCK * ITEMS) + (uint32_t)tid;

#pragma unroll
    for (int k = 0; k < ITEMS; ++k, idx += BLOCK) {
        if (idx >= N_VEC4) break;

        vf4 v = __builtin_nontemporal_load(&x[idx]);   // streamed once: NT

        uint32_t f4  = idx & 31u;           // 32 float4 per (b,c,t) row
        uint32_t row = idx >> 5;            // b*C*T + c*T + t
        uint32_t t   = row % (uint32_t)DIM_T;
        uint32_t c   = (row / (uint32_t)DIM_T) % (uint32_t)DIM_C;
        bool     ch  = (c < (uint32_t)CM);  // last 3 channels never masked
        uint32_t blk = t / (uint32_t)TIME_MASK_STEP;
        int      tin = (int)(t - blk * (uint32_t)TIME_MASK_STEP);

        uint32_t ridx = ch ? (c * (uint32_t)NB + blk) : 0u;
        Rect r = s_rect[ridx];              // wave-uniform LDS broadcast

        bool rowm = apply && ch && (tin >= r.ts) && (tin < r.te);
        if (rowm) {                         // wave-coherent branch
            int f0 = (int)(f4 << 2);
            v.x = (f0 + 0 >= r.fs && f0 + 0 < r.fe) ? LOG_EPS : v.x;
            v.y = (f0 + 1 >= r.fs && f0 + 1 < r.fe) ? LOG_EPS : v.y;
            v.z = (f0 + 2 >= r.fs && f0 + 2 < r.fe) ? LOG_EPS : v.z;
            v.w = (f0 + 3 >= r.fs && f0 + 3 < r.fe) ? LOG_EPS : v.w;
        }
        __builtin_nontemporal_store(v, &out[idx]);     // streamed once: NT
    }
}

// ---------------- host entry ----------------
extern "C" void kernel_launch(void* const* d_in, const int* in_sizes, int n_in,
                              void* d_out, int out_size, void* d_ws, size_t ws_size,
                              hipStream_t stream) {
    const float* x = (const float*)d_in[0];
    float* out = (float*)d_out;
    int* ws = (int*)d_ws;   // 96 * int4 rects (1536 B) + 1 int apply flag

    hipLaunchKernelGGL(rng_setup_kernel, dim3(1), dim3(128), 0, stream, ws);

    const uint32_t per_block = BLOCK * ITEMS;                      // 1024 float4
    const uint32_t grid = (N_VEC4 + per_block - 1u) / per_block;   // 6720 blocks
    hipLaunchKernelGGL(cutoff_apply_kernel, dim3(grid), dim3(BLOCK), 0, stream,
                       (const vf4*)x, (vf4*)out, (const int*)ws);
}